// SinkhornLoss_86517821212127
// MI455X (gfx1250) — compile-verified
//
#include <hip/hip_runtime.h>
#include <hip/hip_bf16.h>

typedef __attribute__((ext_vector_type(2))) float v2f;
typedef __attribute__((ext_vector_type(8))) float v8f;

#define BATCH   128
#define HW      24
#define NPTS    576          // 24*24
#define PAD     32           // padded matmul dim
#define EPS     0.01f
#define IEPS    100.0f       // 1/EPS
#define NITER   5

// D = A * B for 32x32 f32 operands in LDS, single wave32, EXEC all-ones.
// Uses V_WMMA_F32_16X16X4_F32: A 16x4 (v2f/lane), B 4x16 (v2f/lane), C/D 16x16 (v8f/lane).
// A layout: lane L: m=L&15, k0=(L<16)?0:2 -> a[0]=A[m][k0], a[1]=A[m][k0+1]
// B layout: lane L: n=L&15, k0=(L<16)?0:2 -> b[0]=B[k0][n], b[1]=B[k0+1][n]
// D layout: vgpr r: lanes0-15 -> M=r, lanes16-31 -> M=8+r, N=L&15
__device__ __forceinline__ void wave_matmul32(const float* A, const float* B,
                                              float* D, int lane) {
    const int half = lane >> 4;      // 0 or 1
    const int l16  = lane & 15;
    const int kofs = half * 2;
#pragma unroll
    for (int ti = 0; ti < PAD; ti += 16) {
#pragma unroll
        for (int tj = 0; tj < PAD; tj += 16) {
            v8f acc = {};
#pragma unroll
            for (int k0 = 0; k0 < PAD; k0 += 4) {
                const int ka = k0 + kofs;
                v2f a, b;
                a[0] = A[(ti + l16) * PAD + ka];
                a[1] = A[(ti + l16) * PAD + ka + 1];
                b[0] = B[ka * PAD + tj + l16];
                b[1] = B[(ka + 1) * PAD + tj + l16];
                acc = __builtin_amdgcn_wmma_f32_16x16x4_f32(
                    false, a, false, b, (short)0, acc, false, false);
            }
#pragma unroll
            for (int r = 0; r < 8; ++r) {
                const int row = ti + r + half * 8;
                D[row * PAD + tj + l16] = acc[r];
            }
        }
    }
}

// dst[32x32] = exp(src[24x24] * IEPS), zero-padded
__device__ __forceinline__ void pad_exp(float* dst, const float* src, int lane) {
    for (int p = lane; p < PAD * PAD; p += 32) {
        const int r = p >> 5, c = p & 31;
        float val = 0.0f;
        if (r < HW && c < HW) val = __expf(src[r * HW + c] * IEPS);
        dst[p] = val;
    }
}

__device__ __forceinline__ float wave_reduce(float x, float* red, int lane) {
    red[lane] = x;
    __syncthreads();
#pragma unroll
    for (int off = 16; off > 0; off >>= 1) {
        if (lane < off) red[lane] += red[lane + off];
        __syncthreads();
    }
    const float r = red[0];
    __syncthreads();
    return r;
}

__global__ __launch_bounds__(32)
void SinkhornLoss_sep_wmma_kernel(const float* __restrict__ canvas,
                                  const float* __restrict__ gt,
                                  float* __restrict__ part) {
    __shared__ float lm[NPTS], ln[NPTS], u[NPTS], v[NPTS];
    __shared__ float K32[PAD * PAD], Kc32[PAD * PAD];
    __shared__ float S0[PAD * PAD], S1[PAD * PAD], S2[PAD * PAD];
    __shared__ float red[32];

    const int b    = blockIdx.x;
    const int lane = threadIdx.x;

    // ---- 4x4 area pool of channel 0, masses into u (canvas) / v (gt) ----
    const float* cbase = canvas + (size_t)b * 3 * 96 * 96;   // channel 0
    const float* gbase = gt     + (size_t)b * 3 * 96 * 96;
    for (int o = lane; o < NPTS; o += 32) {
        const int oy = o / HW, ox = o % HW;
        float sc = 0.0f, sg = 0.0f;
#pragma unroll
        for (int r = 0; r < 4; ++r) {
            const float4 c4 = *(const float4*)(cbase + (oy * 4 + r) * 96 + ox * 4);
            const float4 g4 = *(const float4*)(gbase + (oy * 4 + r) * 96 + ox * 4);
            sc += c4.x + c4.y + c4.z + c4.w;
            sg += g4.x + g4.y + g4.z + g4.w;
        }
        float mx = sc * (1.0f / 16.0f);
        float my = sg * (1.0f / 16.0f);
        mx = fminf(fmaxf(mx, 0.0f), 1e9f) + 1e-9f;
        my = fminf(fmaxf(my, 0.0f), 1e9f) + 1e-9f;
        u[o] = mx;
        v[o] = my;
    }
    __syncthreads();

    // ---- normalize -> log_mu / log_nu ----
    float sx = 0.0f, sy = 0.0f;
    for (int o = lane; o < NPTS; o += 32) { sx += u[o]; sy += v[o]; }
    const float tot_x = wave_reduce(sx, red, lane);
    const float tot_y = wave_reduce(sy, red, lane);
    for (int o = lane; o < NPTS; o += 32) {
        lm[o] = __logf(u[o] / tot_x);
        ln[o] = __logf(v[o] / tot_y);
    }
    __syncthreads();
    for (int o = lane; o < NPTS; o += 32) { u[o] = 0.0f; v[o] = 0.0f; }

    // ---- separable Gaussian kernels K, Kc (zero-padded to 32x32) ----
    for (int p = lane; p < PAD * PAD; p += 32) {
        const int a = p >> 5, c = p & 31;
        float kv = 0.0f, kc = 0.0f;
        if (a < HW && c < HW) {
            const float d  = (float)(a - c) * (1.0f / 24.0f);
            const float d2 = d * d;
            kv = __expf(-d2 * IEPS);
            kc = d2 * kv;
        }
        K32[p]  = kv;
        Kc32[p] = kc;
    }
    __syncthreads();

    // ---- Sinkhorn iterations: u = eps*(log_mu - log(K*exp(v/eps)*K)), etc. ----
    for (int it = 0; it < NITER; ++it) {
        pad_exp(S0, v, lane);                     __syncthreads();
        wave_matmul32(K32, S0, S1, lane);         __syncthreads();
        wave_matmul32(S1, K32, S2, lane);         __syncthreads();
        for (int o = lane; o < NPTS; o += 32)
            u[o] = EPS * (lm[o] - __logf(S2[(o / HW) * PAD + (o % HW)]));
        __syncthreads();

        pad_exp(S0, u, lane);                     __syncthreads();
        wave_matmul32(K32, S0, S1, lane);         __syncthreads();
        wave_matmul32(S1, K32, S2, lane);         __syncthreads();
        for (int o = lane; o < NPTS; o += 32)
            v[o] = EPS * (ln[o] - __logf(S2[(o / HW) * PAD + (o % HW)]));
        __syncthreads();
    }

    // ---- cost = sum A .* (Kc*B*K + K*B*Kc),  A=exp(u/eps), B=exp(v/eps) ----
    pad_exp(S0, v, lane);                         __syncthreads();
    wave_matmul32(Kc32, S0, S1, lane);            __syncthreads();
    wave_matmul32(S1, K32, S2, lane);             __syncthreads();  // term1 -> S2
    wave_matmul32(K32, S0, S1, lane);             __syncthreads();
    wave_matmul32(S1, Kc32, S0, lane);            __syncthreads();  // term2 -> S0

    float cacc = 0.0f;
    for (int o = lane; o < NPTS; o += 32) {
        const int idx = (o / HW) * PAD + (o % HW);
        cacc += __expf(u[o] * IEPS) * (S2[idx] + S0[idx]);
    }
    const float cost = wave_reduce(cacc, red, lane);
    if (lane == 0) part[b] = cost;
}

// Fixed-order reduction -> deterministic scalar mean.
__global__ __launch_bounds__(32)
void SinkhornLoss_final_reduce(const float* __restrict__ part,
                               float* __restrict__ out) {
    if (threadIdx.x == 0 && blockIdx.x == 0) {
        float s = 0.0f;
        for (int i = 0; i < BATCH; ++i) s += part[i];
        out[0] = s * (1.0f / (float)BATCH);
    }
}

extern "C" void kernel_launch(void* const* d_in, const int* in_sizes, int n_in,
                              void* d_out, int out_size, void* d_ws, size_t ws_size,
                              hipStream_t stream) {
    const float* canvas = (const float*)d_in[0];
    const float* gt     = (const float*)d_in[1];
    float* part = (float*)d_ws;                  // 128 floats of scratch
    float* out  = (float*)d_out;

    SinkhornLoss_sep_wmma_kernel<<<BATCH, 32, 0, stream>>>(canvas, gt, part);
    SinkhornLoss_final_reduce<<<1, 32, 0, stream>>>(part, out);
}